// EMDHybridInSARModel_85779086835986
// MI455X (gfx1250) — compile-verified
//
#include <hip/hip_runtime.h>
#include <hip/hip_bf16.h>
#include <math.h>

typedef __attribute__((ext_vector_type(2))) float v2f;
typedef __attribute__((ext_vector_type(8))) float v8f;

#define SMOOTH_F 0.15f
#define MAXK 8   // compute unroll width; runtime K<=MAXK handled by zero-fill

// ---------------------------------------------------------------------------
// K1: per-station coefficient build.
// P[n][0]=offset, P[n][1]=trend, P[n][2+2c]=amp_s*cos(phi_s), P[n][3+2c]=amp_s*sin(phi_s)
// cos/sin of atan2(im,re) computed as re/h, im/h (no atan2 needed).
// Phases are in [0, 2*pi) so hardware __sinf/__cosf are accurate here.
// ---------------------------------------------------------------------------
__global__ void build_coeffs(const float* __restrict__ off,
                             const float* __restrict__ tr,
                             const float* __restrict__ amp,
                             const float* __restrict__ ph,
                             const float* __restrict__ sw,
                             const float* __restrict__ w,
                             const int*   __restrict__ idx,
                             float* __restrict__ P,
                             float* __restrict__ mix,
                             int N, int Np, int K, int C)
{
    int n = blockIdx.x * blockDim.x + threadIdx.x;
    if (n >= Np) return;
    float* p = P + (unsigned)n * 8u;
    if (n >= N) {   // zero-pad rows so WMMA A-loads on the padded tile are benign
        #pragma unroll
        for (int i = 0; i < 8; ++i) p[i] = 0.0f;
        return;
    }
    p[0] = off[n];
    p[1] = tr[n];
    mix[n] = 1.0f / (1.0f + __expf(-sw[n]));

    float aa[3] = {0.f, 0.f, 0.f};
    float ac[3] = {0.f, 0.f, 0.f};
    float as[3] = {0.f, 0.f, 0.f};
    for (int k = 0; k < K; ++k) {
        int   j  = idx[(unsigned)n * K + k];
        float wk = w  [(unsigned)n * K + k];
        for (int c = 0; c < C && c < 3; ++c) {
            float av = amp[(unsigned)j * C + c];
            float pv = ph [(unsigned)j * C + c];
            aa[c] += wk * av;
            ac[c] += wk * __cosf(pv);
            as[c] += wk * __sinf(pv);
        }
    }
    for (int c = 0; c < C && c < 3; ++c) {
        float a0   = amp[(unsigned)n * C + c];
        float p0   = ph [(unsigned)n * C + c];
        float amps = (1.0f - SMOOTH_F) * a0 + SMOOTH_F * aa[c];
        float re   = (1.0f - SMOOTH_F) * __cosf(p0) + SMOOTH_F * ac[c];
        float im   = (1.0f - SMOOTH_F) * __sinf(p0) + SMOOTH_F * as[c];
        float h2   = re * re + im * im;
        float cp, sp;
        if (h2 > 1e-30f) {
            float rh = rsqrtf(h2);
            cp = re * rh; sp = im * rh;
        } else {            // atan2(0,0)==0 -> cos=1, sin=0
            cp = 1.0f; sp = 0.0f;
        }
        p[2 + 2 * c] = amps * cp;
        p[3 + 2 * c] = amps * sp;
    }
}

// ---------------------------------------------------------------------------
// K2: time basis Q[8][Tp] = {1, t, sin(w0 t), cos(w0 t), ...}, zero pad cols.
// Arguments reach ~1900 rad -> use precise libm sinf/cosf (only Tp threads).
// ---------------------------------------------------------------------------
__global__ void build_basis(const float* __restrict__ time,
                            const float* __restrict__ periods,
                            float* __restrict__ Q,
                            int T, int Tp, int C)
{
    int t = blockIdx.x * blockDim.x + threadIdx.x;
    if (t >= Tp) return;
    if (t >= T) {
        #pragma unroll
        for (int r = 0; r < 8; ++r) Q[(unsigned)r * Tp + t] = 0.0f;
        return;
    }
    float tv = time[t];
    Q[(unsigned)0 * Tp + t] = 1.0f;
    Q[(unsigned)1 * Tp + t] = tv;
    for (int c = 0; c < C && c < 3; ++c) {
        float om  = 6.2831853071795864f / periods[c];
        float ang = om * tv;
        Q[(unsigned)(2 + 2 * c) * Tp + t] = sinf(ang);
        Q[(unsigned)(3 + 2 * c) * Tp + t] = cosf(ang);
    }
}

// ---------------------------------------------------------------------------
// K3: emd[n][t] = sum_{j<4} emd_seasonal[n][j][t], zero-padded to [Np][Tp].
// Vectorized float4 path (biggest HBM pass: 240 MB read + 60 MB write).
// ---------------------------------------------------------------------------
__global__ void build_emd_vec4(const float* __restrict__ es,
                               float* __restrict__ emd,
                               int N, int T, int Tp)
{
    int t4 = (blockIdx.x * blockDim.x + threadIdx.x) * 4;  // Tp % 4 == 0
    int n  = blockIdx.y;
    if (t4 >= Tp) return;
    float4 v = make_float4(0.f, 0.f, 0.f, 0.f);
    if (n < N && t4 + 3 < T) {
        const float4* b0 = (const float4*)(es + (unsigned)n * 4u * (unsigned)T + t4);
        unsigned ts = (unsigned)T / 4u;  // float4 stride between components
        float4 c0 = b0[0], c1 = b0[ts], c2 = b0[2 * ts], c3 = b0[3 * ts];
        v.x = c0.x + c1.x + c2.x + c3.x;
        v.y = c0.y + c1.y + c2.y + c3.y;
        v.z = c0.z + c1.z + c2.z + c3.z;
        v.w = c0.w + c1.w + c2.w + c3.w;
    }
    *(float4*)(emd + (unsigned)n * (unsigned)Tp + t4) = v;
}

__global__ void build_emd_scalar(const float* __restrict__ es,
                                 float* __restrict__ emd,
                                 int N, int T, int Tp)
{
    int t = blockIdx.x * blockDim.x + threadIdx.x;
    int n = blockIdx.y;
    if (t >= Tp) return;
    float v = 0.0f;
    if (n < N && t < T) {
        const float* b = es + (unsigned)n * 4u * (unsigned)T + t;
        v = b[0] + b[(unsigned)T] + b[(unsigned)2 * T] + b[(unsigned)3 * T];
    }
    emd[(unsigned)n * (unsigned)Tp + t] = v;
}

// ---------------------------------------------------------------------------
// K4: fused tile kernel. One wave per 16-station tile, strip-mined over time
// tiles. Tile-invariant data staged once in LDS (neighbor row offsets are
// pre-multiplied by Tp and zero-filled for k>=K / stations>=N, so the compute
// loop is a branch-free fully-unrolled 8-term FMA chain). All epilogue loads
// are unguarded (every buffer is zero-padded); only the final store carries a
// predicate, which is uniformly true on interior tiles.
// C/D layout: VGPR v -> station (v + 8*(lane>=16)), N col = lane&15.
// A layout:   lanes 0-15: V0=K0,V1=K1 ; lanes 16-31: V0=K2,V1=K3 (M=lane&15).
// ---------------------------------------------------------------------------
__global__ __launch_bounds__(32)
void fuse_tile(const float* __restrict__ P,
               const float* __restrict__ Q,
               const float* __restrict__ emd,
               const float* __restrict__ mix,
               const float* __restrict__ w,
               const int*   __restrict__ idx,
               float* __restrict__ out,
               int N, int T, int Tp, int K)
{
    __shared__ unsigned s_row[16 * MAXK];  // idx[n,k] * Tp (element offset), 0 if unused
    __shared__ float    s_w  [16 * MAXK];  // weight, 0 if unused
    __shared__ float    s_mix[16];
    __shared__ float    s_P  [16 * 8];

    const int lane = threadIdx.x;
    const int half = lane >> 4;
    const int lp   = lane & 15;
    const int n0   = blockIdx.x * 16;

    // ---- stage tile-invariant data into LDS ----
    for (int e = lane; e < 16 * 8; e += 32)
        s_P[e] = P[(unsigned)n0 * 8u + e];        // P is padded to Np rows
    #pragma unroll
    for (int e0 = 0; e0 < 16 * MAXK; e0 += 32) {
        int e = e0 + lane;
        int r = e >> 3;        // station row in tile (MAXK == 8)
        int k = e & 7;         // neighbor slot
        unsigned row = 0u;
        float    wk  = 0.0f;
        if (k < K && (n0 + r) < N) {
            unsigned src = (unsigned)(n0 + r) * (unsigned)K + (unsigned)k;
            row = (unsigned)idx[src] * (unsigned)Tp;
            wk  = w[src];
        }
        s_row[e] = row;
        s_w[e]   = wk;
    }
    if (lane < 16) {
        int sg = n0 + lane;
        s_mix[lane] = (sg < N) ? mix[sg] : 0.0f;
    }
    __syncthreads();

    // ---- A fragments (constant over the time loop) ----
    v2f a0, a1;
    a0.x = s_P[lp * 8 + half * 2 + 0];
    a0.y = s_P[lp * 8 + half * 2 + 1];
    a1.x = s_P[lp * 8 + half * 2 + 4];
    a1.y = s_P[lp * 8 + half * 2 + 5];

    const bool fullN   = (n0 + 16 <= N);
    const int  nTilesT = Tp >> 4;

    for (int tt = blockIdx.y; tt < nTilesT; tt += gridDim.y) {
        const int t0 = tt << 4;
        const int t  = t0 + lp;

        v2f b0, b1;
        b0.x = Q[(unsigned)(half * 2 + 0) * (unsigned)Tp + t];
        b0.y = Q[(unsigned)(half * 2 + 1) * (unsigned)Tp + t];
        b1.x = Q[(unsigned)(4 + half * 2 + 0) * (unsigned)Tp + t];
        b1.y = Q[(unsigned)(4 + half * 2 + 1) * (unsigned)Tp + t];

        v8f c = {};
        c = __builtin_amdgcn_wmma_f32_16x16x4_f32(false, a0, false, b0, (short)0, c, false, false);
        c = __builtin_amdgcn_wmma_f32_16x16x4_f32(false, a1, false, b1, (short)0, c, false, false);

        const bool full = fullN & (t0 + 16 <= T);   // uniform: whole tile valid

        #pragma unroll
        for (int v = 0; v < 8; ++v) {
            const int r  = v + half * 8;            // station row within tile
            const int sg = n0 + r;
            // All loads are safe: emd/mix/row tables zero-padded.
            float m = s_mix[r];
            float e = emd[(unsigned)sg * (unsigned)Tp + (unsigned)t];
            float g = 0.0f;
            #pragma unroll
            for (int k = 0; k < MAXK; ++k)
                g += s_w[r * MAXK + k] * emd[s_row[r * MAXK + k] + (unsigned)t];
            float val = c[v] + (1.0f - m) * e + m * g;
            if (full | ((sg < N) & (t < T)))
                out[(unsigned)sg * (unsigned)T + (unsigned)t] = val;
        }
    }
}

// ---------------------------------------------------------------------------
extern "C" void kernel_launch(void* const* d_in, const int* in_sizes, int n_in,
                              void* d_out, int out_size, void* d_ws, size_t ws_size,
                              hipStream_t stream)
{
    const float* time = (const float*)d_in[0];   // [T]
    const float* off  = (const float*)d_in[1];   // [N]
    const float* tr   = (const float*)d_in[2];   // [N]
    const float* es   = (const float*)d_in[3];   // [N,4,T]
    const float* amp  = (const float*)d_in[4];   // [N,C]
    const float* ph   = (const float*)d_in[5];   // [N,C]
    const float* per  = (const float*)d_in[6];   // [C]
    const float* sw   = (const float*)d_in[7];   // [N]
    const float* w    = (const float*)d_in[8];   // [N,K]
    const int*   idx  = (const int*)  d_in[9];   // [N,K]

    int T  = in_sizes[0];
    int N  = in_sizes[1];
    int C  = in_sizes[6];
    int K  = in_sizes[8] / N;
    if (K > MAXK) K = MAXK;
    int Tp = (T + 15) & ~15;
    int Np = (N + 15) & ~15;

    // Workspace layout (256B-aligned regions)
    char*  ws = (char*)d_ws;
    size_t o  = 0;
    float* P   = (float*)(ws + o); o += (size_t)Np * 8 * sizeof(float); o = (o + 255) & ~(size_t)255;
    float* mix = (float*)(ws + o); o += (size_t)Np * sizeof(float);     o = (o + 255) & ~(size_t)255;
    float* Q   = (float*)(ws + o); o += (size_t)8 * Tp * sizeof(float); o = (o + 255) & ~(size_t)255;
    float* emd = (float*)(ws + o); o += (size_t)Np * Tp * sizeof(float);
    float* out = (float*)d_out;

    build_coeffs<<<dim3((Np + 255) / 256), dim3(256), 0, stream>>>(
        off, tr, amp, ph, sw, w, idx, P, mix, N, Np, K, C);

    build_basis<<<dim3((Tp + 255) / 256), dim3(256), 0, stream>>>(
        time, per, Q, T, Tp, C);

    if ((T & 3) == 0) {
        build_emd_vec4<<<dim3((Tp / 4 + 255) / 256, Np), dim3(256), 0, stream>>>(
            es, emd, N, T, Tp);
    } else {
        build_emd_scalar<<<dim3((Tp + 255) / 256, Np), dim3(256), 0, stream>>>(
            es, emd, N, T, Tp);
    }

    int nTilesT = Tp / 16;
    int strips  = nTilesT < 16 ? nTilesT : 16;
    fuse_tile<<<dim3(Np / 16, strips), dim3(32), 0, stream>>>(
        P, Q, emd, mix, w, idx, out, N, T, Tp, K);
}